// DMargin_11940009083628
// MI455X (gfx1250) — compile-verified
//
#include <hip/hip_runtime.h>
#include <hip/hip_bf16.h>

typedef __attribute__((ext_vector_type(2))) float v2f;
typedef __attribute__((ext_vector_type(8))) float v8f;

#define B_ROWS 4096
#define D_K    512
#define C_CLS  10575
#define ROWS   32                       // rows of B per workgroup (2 x 16-row WMMA tiles)
#define WAVES  16                       // waves per workgroup (512 threads)
#define ASTRIDE 516                     // padded LDS row stride (dwords): bank step 4 -> conflict-free b64 reads
#define NTILE  ((C_CLS + 15) / 16)      // 661 column tiles (last one partial: 15 cols)

// ---------------------------------------------------------------------------
// Kernel A: per-sample prep. One wave per sample i:
//   orig_t[i] = dot(feats[i], w[:, labels[i]])
//   dist[i]   = sum_k (mn_k)^2, mn = (centers[l]/counts[l]) / ||centers[l]/counts[l]||
// Writes ws[2i] = orig_t, ws[2i+1] = orig_t + dist. Also zeroes d_out.
// ---------------------------------------------------------------------------
__global__ __launch_bounds__(128) void dmargin_prep(
    const float* __restrict__ feats, const int* __restrict__ labels,
    const float* __restrict__ centers, const float* __restrict__ counts,
    const float* __restrict__ w, float* __restrict__ tvals, float* __restrict__ out)
{
    if (blockIdx.x == 0 && threadIdx.x == 0) out[0] = 0.0f;  // d_out is poisoned; zero for atomics

    const int lane = threadIdx.x & 31;
    const int i    = blockIdx.x * 4 + (threadIdx.x >> 5);
    const int l    = labels[i];
    const float cnt = counts[l];

    const float* frow = feats   + (size_t)i * D_K;
    const float* crow = centers + (size_t)l * D_K;
    const float* wcol = w + l;

    float acc = 0.0f, ssum = 0.0f;
    float mv[16];
#pragma unroll
    for (int j = 0; j < 16; ++j) {
        const int k = lane + 32 * j;
        acc += frow[k] * wcol[k * C_CLS];
        const float mk = crow[k] / cnt;
        mv[j] = mk;
        ssum += mk * mk;
    }
#pragma unroll
    for (int off = 1; off < 32; off <<= 1) {
        acc  += __shfl_xor(acc,  off, 32);
        ssum += __shfl_xor(ssum, off, 32);
    }
    const float norm = sqrtf(ssum);
    float dot = 0.0f;
#pragma unroll
    for (int j = 0; j < 16; ++j) { const float u = mv[j] / norm; dot += u * u; }
#pragma unroll
    for (int off = 1; off < 32; off <<= 1) dot += __shfl_xor(dot, off, 32);

    if (lane == 0) {
        tvals[2 * i]     = acc;        // original[i, l]
        tvals[2 * i + 1] = acc + dot;  // true target logit
    }
}

// ---------------------------------------------------------------------------
// Kernel B: fused GEMM (fp32 WMMA 16x16x4) + streaming log-softmax NLL.
// Block = 32 rows (two 16-row WMMA tiles sharing every B fragment) x all C.
// Each of 16 waves sweeps column tiles stride-16 with per-row running
// (max, sumexp) in registers (online softmax); logits never touch HBM.
// ---------------------------------------------------------------------------
__global__ __launch_bounds__(512) void dmargin_fused(
    const float* __restrict__ feats, const float* __restrict__ w,
    const float* __restrict__ tvals, float* __restrict__ out)
{
    __shared__ float Atile[ROWS * ASTRIDE];     // 32 x 512 fp32, padded
    __shared__ float redM[WAVES][ROWS];
    __shared__ float redS[WAVES][ROWS];

    const int tid     = threadIdx.x;
    const int rowbase = blockIdx.x * ROWS;

    // Stage A (32 x 512 fp32) into LDS, float4-coalesced: 4096 float4 / 512 thr = 8 each.
    {
        const float4* src = (const float4*)(feats + (size_t)rowbase * D_K);
#pragma unroll
        for (int p = 0; p < (ROWS * D_K / 4) / 512; ++p) {
            const int idx4 = tid + p * 512;
            const int e    = idx4 * 4;          // dword index in packed tile
            const int row  = e >> 9;            // /512
            const int col  = e & (D_K - 1);
            *(float4*)&Atile[row * ASTRIDE + col] = src[idx4];
        }
    }
    __syncthreads();

    const int wave = tid >> 5;
    const int lane = tid & 31;
    const int hi   = lane >> 4;   // half-wave select
    const int ln   = lane & 15;

    float m0[8], s0[8], m1[8], s1[8];
#pragma unroll
    for (int v = 0; v < 8; ++v) {
        m0[v] = -1.0e30f; s0[v] = 0.0f;
        m1[v] = -1.0e30f; s1[v] = 0.0f;
    }

    for (int t = wave; t < NTILE; t += WAVES) {
        const int cbase  = t * 16;
        const int c      = cbase + ln;
        const int cclamp = (c < C_CLS) ? c : (C_CLS - 1);
        const float* wp  = w + cclamp;

        v8f acc0 = {};
        v8f acc1 = {};
        const int abase0 = ln * ASTRIDE + 2 * hi;           // A frag, row tile 0 (M=ln)
        const int abase1 = (ln + 16) * ASTRIDE + 2 * hi;    // A frag, row tile 1 (M=ln+16)
#pragma unroll 4
        for (int k = 0; k < D_K; k += 4) {
            const float2 a0 = *(const float2*)&Atile[abase0 + k];  // ds_load_b64 (2addr-merged)
            const float2 a1 = *(const float2*)&Atile[abase1 + k];
            const int kb = k + 2 * hi;                             // B frag: VGPR j -> K = kb + j
            v2f bv; bv.x = wp[kb * C_CLS]; bv.y = wp[(kb + 1) * C_CLS];
            v2f av0; av0.x = a0.x; av0.y = a0.y;
            v2f av1; av1.x = a1.x; av1.y = a1.y;
            acc0 = __builtin_amdgcn_wmma_f32_16x16x4_f32(
                       false, av0, false, bv, (short)0, acc0, false, false);
            acc1 = __builtin_amdgcn_wmma_f32_16x16x4_f32(
                       false, av1, false, bv, (short)0, acc1, false, false);
        }

        const float ok = (c < C_CLS) ? 1.0f : 0.0f;   // mask the 1 fake column of the last tile
#pragma unroll
        for (int v = 0; v < 8; ++v) {
            float val = ok ? acc0[v] : -1.0e30f;
            float mn2 = fmaxf(m0[v], val);
            s0[v] = s0[v] * __expf(m0[v] - mn2) + ok * __expf(val - mn2);
            m0[v] = mn2;

            val = ok ? acc1[v] : -1.0e30f;
            mn2 = fmaxf(m1[v], val);
            s1[v] = s1[v] * __expf(m1[v] - mn2) + ok * __expf(val - mn2);
            m1[v] = mn2;
        }
    }

    // Reduce (max, sumexp) across the 16 lanes holding the same row (xor stays in-half).
#pragma unroll
    for (int g = 0; g < 2; ++g) {
#pragma unroll
        for (int v = 0; v < 8; ++v) {
            float mm = g ? m1[v] : m0[v];
            float ss = g ? s1[v] : s0[v];
#pragma unroll
            for (int off = 1; off < 16; off <<= 1) {
                const float mo = __shfl_xor(mm, off, 32);
                const float so = __shfl_xor(ss, off, 32);
                const float mn2 = fmaxf(mm, mo);
                ss = ss * __expf(mm - mn2) + so * __expf(mo - mn2);
                mm = mn2;
            }
            if (ln == 0) {
                const int r = v + 8 * hi + 16 * g;    // row within block (D-matrix layout)
                redM[wave][r] = mm;
                redS[wave][r] = ss;
            }
        }
    }
    __syncthreads();

    // Combine the 16 waves' disjoint column ranges, apply target correction, emit NLL.
    if (tid < ROWS) {
        float mm = redM[0][tid], ss = redS[0][tid];
#pragma unroll
        for (int wv = 1; wv < WAVES; ++wv) {
            const float mo = redM[wv][tid], so = redS[wv][tid];
            const float mn2 = fmaxf(mm, mo);
            ss = ss * __expf(mm - mn2) + so * __expf(mo - mn2);
            mm = mn2;
        }
        const int row = rowbase + tid;
        const float ot = tvals[2 * row];      // original[i, l] (already inside mm/ss)
        const float tt = tvals[2 * row + 1];  // original + dist
        const float m2 = fmaxf(mm, tt);
        const float s2 = ss * __expf(mm - m2) - __expf(ot - m2) + __expf(tt - m2);
        const float nll = m2 + __logf(s2) - tt;
        atomicAdd(out, nll * (1.0f / (float)B_ROWS));
    }
}

extern "C" void kernel_launch(void* const* d_in, const int* in_sizes, int n_in,
                              void* d_out, int out_size, void* d_ws, size_t ws_size,
                              hipStream_t stream) {
    (void)in_sizes; (void)n_in; (void)out_size; (void)ws_size;
    const float* feats   = (const float*)d_in[0];
    const int*   labels  = (const int*)  d_in[1];
    const float* centers = (const float*)d_in[2];
    const float* counts  = (const float*)d_in[3];
    const float* w       = (const float*)d_in[4];
    float*       out     = (float*)d_out;
    float*       tvals   = (float*)d_ws;   // 4096 x {orig_t, orig_t+dist} = 32 KB

    dmargin_prep<<<B_ROWS / 4, 128, 0, stream>>>(feats, labels, centers, counts, w, tvals, out);
    dmargin_fused<<<B_ROWS / ROWS, 512, 0, stream>>>(feats, w, tvals, out);
}